// KGEncoder_84121229460003
// MI455X (gfx1250) — compile-verified
//
#include <hip/hip_runtime.h>
#include <hip/hip_bf16.h>
#include <math.h>

// Problem constants (match reference)
#define Nn    40960
#define Ee    655360
#define Dd    128
#define Gg    256
#define Qq    3
#define Kk    512
#define FOUTc 512
#define ENe   (Ee + Nn)
#define LDA   132   // padded LDS row stride (floats): conflict-free for 16x16x4 frag reads

typedef __attribute__((ext_vector_type(2))) float v2f;
typedef __attribute__((ext_vector_type(8))) float v8f;
typedef __attribute__((ext_vector_type(4))) int   v4i;

// pointer-to-int4 in global (AS1) / LDS (AS3) address spaces
typedef __attribute__((address_space(1))) v4i* gv4ptr;
typedef __attribute__((address_space(3))) v4i* lv4ptr;

// --------------------------- CDNA5 async-to-LDS ----------------------------
#if defined(__has_builtin)
#  if __has_builtin(__builtin_amdgcn_global_load_async_to_lds_b128)
#    define HAVE_ASYNC 1
#  else
#    define HAVE_ASYNC 0
#  endif
#else
#  define HAVE_ASYNC 0
#endif

__device__ inline void wait_ds0() {
#if defined(__has_builtin) && __has_builtin(__builtin_amdgcn_s_wait_dscnt)
    __builtin_amdgcn_s_wait_dscnt(0);
#else
    asm volatile("s_wait_dscnt 0" ::: "memory");
#endif
}

template <int N> __device__ inline void wait_async_le() {
#if defined(__has_builtin) && __has_builtin(__builtin_amdgcn_s_wait_asynccnt)
    __builtin_amdgcn_s_wait_asynccnt(N);
#else
    asm volatile("s_wait_asynccnt %0" ::"n"(N) : "memory");
#endif
}

// Move one 128-float row (512B) HBM -> LDS: 32 lanes x b128, ASYNCcnt-tracked.
__device__ inline void tile_row_async(const float* __restrict__ grow, float* lrow, int lane) {
#if HAVE_ASYNC
    __builtin_amdgcn_global_load_async_to_lds_b128(
        (gv4ptr)(grow + lane * 4),
        (lv4ptr)(lrow + lane * 4), 0, 0);
#else
    *(float4*)&lrow[lane * 4] = *(const float4*)&grow[lane * 4];
#endif
}

// ---------------------------------------------------------------------------
// float atomic max via integer bit tricks (init must be 0xFF800000 = -inf)
__device__ inline void atomicMaxFloat(float* addr, float v) {
    if (v >= 0.0f) atomicMax((int*)addr, __float_as_int(v));
    else           atomicMin((unsigned int*)addr, __float_as_uint(v));
}

__global__ void fill_neginf_kernel(float* p, int n) {
    int i = blockIdx.x * blockDim.x + threadIdx.x;
    if (i < n) p[i] = __int_as_float(0xFF800000);
}

// ---------------------------------------------------------------------------
// Self-loop attr: sum incoming edge_attr per target + degree
__global__ void deg_loop_kernel(const int* __restrict__ eidx,
                                const float* __restrict__ eattr,
                                float* __restrict__ lattr, float* __restrict__ deg) {
    int e = blockIdx.x;           // one block per edge
    int d = threadIdx.x;          // 128 threads
    int t = eidx[Ee + e];
    atomicAdd(&lattr[(long)t * Dd + d], eattr[(long)e * Dd + d]);
    if (d == 0) atomicAdd(&deg[t], 1.0f);
}

__global__ void loop_norm_kernel(float* __restrict__ lattr, const float* __restrict__ deg) {
    long i = (long)blockIdx.x * blockDim.x + threadIdx.x;
    if (i < (long)Nn * Dd) {
        int n = (int)(i >> 7);
        lattr[i] /= fmaxf(deg[n], 1.0f);
    }
}

// ---------------------------------------------------------------------------
// Generic WMMA GEMM:  C[M, Dout] = A[M,128] @ W[Dout,128]^T + bias
// block = 32 (1 wave); grid = (M/16, Dout/16). M, Dout divisible by 16.
__global__ void gemm128_wmma(const float* __restrict__ A, const float* __restrict__ W,
                             const float* __restrict__ bias, float* __restrict__ C,
                             int Dout) {
    __shared__ __align__(16) float sA[16 * LDA];
    __shared__ __align__(16) float sB[16 * LDA];
    int lane = threadIdx.x;
    long m0 = (long)blockIdx.x * 16;
    int  n0 = blockIdx.y * 16;
    #pragma unroll
    for (int i = 0; i < 16; i++) {
        int idx4 = lane + i * 32;
        int row = idx4 >> 5, c4 = idx4 & 31;
        *(float4*)&sA[row * LDA + c4 * 4] = *(const float4*)&A[(m0 + row) * Dd + c4 * 4];
        *(float4*)&sB[row * LDA + c4 * 4] = *(const float4*)&W[(long)(n0 + row) * Dd + c4 * 4];
    }
    __syncthreads();
    int l = lane & 15, half = lane >> 4;
    const float* ar = sA + l * LDA + 2 * half;
    const float* br = sB + l * LDA + 2 * half;
    v8f acc = {};
    #pragma unroll
    for (int k = 0; k < Dd; k += 4) {
        v2f a = *(const v2f*)(ar + k);
        v2f b = *(const v2f*)(br + k);
        acc = __builtin_amdgcn_wmma_f32_16x16x4_f32(false, a, false, b, (short)0, acc,
                                                    false, false);
    }
    float bv = bias[n0 + l];
    #pragma unroll
    for (int v = 0; v < 8; v++) {
        long row = m0 + v + 8 * half;
        C[row * Dout + n0 + l] = acc[v] + bv;
    }
}

// ---------------------------------------------------------------------------
// Fused edge kernel (persistent, grid-stride, double-buffered async HBM->LDS):
// per 16-edge tile compute e = eattr @ W_e^T via WMMA with W_e LDS-resident,
// add gathered xl[src] + xr[tgt] (L2-resident), leaky-relu, dot with att,
// half-wave shfl reduce -> logits[E+N].  blockDim = 128 (4 waves).
__global__ void edge_logits_wmma(const float* __restrict__ eattr,
                                 const float* __restrict__ lattr,
                                 const int*  __restrict__ eidx,
                                 const float* __restrict__ We,
                                 const float* __restrict__ att,
                                 const float* __restrict__ xl,
                                 const float* __restrict__ xr,
                                 float* __restrict__ logits) {
    __shared__ __align__(16) float sW[128 * LDA];          // W_e as [o][k], padded
    __shared__ __align__(16) float sA[2 * 4 * 16 * LDA];   // double-buffered per-wave tiles
    int tid = threadIdx.x;          // 0..127
    int w = tid >> 5, lane = tid & 31;
    // Load W_e (128x128) cooperatively: 4096 float4 / 128 threads
    #pragma unroll
    for (int i = 0; i < 32; i++) {
        int idx4 = tid + i * 128;
        int o = idx4 >> 5, c4 = idx4 & 31;
        *(float4*)&sW[o * LDA + c4 * 4] = *(const float4*)&We[(long)o * Dd + c4 * 4];
    }
    __syncthreads();

    const long T = (long)ENe / 64;     // groups of (4 waves x 16 rows)
    const long stride = gridDim.x;
    int l = lane & 15, half = lane >> 4;

    long grp = blockIdx.x;
    // prologue: issue this wave's first tile into buffer 0
    if (grp < T) {
        float* buf = sA + (0 * 4 + w) * 16 * LDA;
        long m0 = (grp * 4 + w) * 16;
        #pragma unroll
        for (int j = 0; j < 16; j++) {
            long r = m0 + j;
            const float* rp = (r < Ee) ? (eattr + r * Dd) : (lattr + (r - Ee) * Dd);
            tile_row_async(rp, buf + j * LDA, lane);
        }
    }
    int p = 0;
    for (; grp < T; grp += stride, p ^= 1) {
        long m0 = (grp * 4 + w) * 16;
        long nxt = grp + stride;
        if (nxt < T) {
#if HAVE_ASYNC
            wait_ds0();                 // buffer 1-p: old LDS reads fully drained
#endif
            float* nbuf = sA + ((1 - p) * 4 + w) * 16 * LDA;
            long nm0 = (nxt * 4 + w) * 16;
            #pragma unroll
            for (int j = 0; j < 16; j++) {
                long r = nm0 + j;
                const float* rp = (r < Ee) ? (eattr + r * Dd) : (lattr + (r - Ee) * Dd);
                tile_row_async(rp, nbuf + j * LDA, lane);
            }
#if HAVE_ASYNC
            wait_async_le<16>();        // in-order: current tile's 16 issues done
#endif
        } else {
#if HAVE_ASYNC
            wait_async_le<0>();
#endif
        }
        // edge endpoints: lane j<16 holds row j's (src,tgt); broadcast via shfl
        long r = m0 + l;
        int s_, t_;
        if (r < Ee) { s_ = eidx[r]; t_ = eidx[Ee + r]; }
        else        { s_ = (int)(r - Ee); t_ = s_; }
        int sIdx[8], tIdx[8];
        float part[8];
        #pragma unroll
        for (int v = 0; v < 8; v++) {
            sIdx[v] = __shfl(s_, v + 8 * half);
            tIdx[v] = __shfl(t_, v + 8 * half);
            part[v] = 0.0f;
        }
        const float* buf = sA + (p * 4 + w) * 16 * LDA;
        const float* ar = buf + l * LDA + 2 * half;
        for (int o0 = 0; o0 < Dd; o0 += 16) {
            const float* br = sW + (o0 + l) * LDA + 2 * half;
            v8f acc = {};
            #pragma unroll
            for (int k = 0; k < Dd; k += 4) {
                v2f a = *(const v2f*)(ar + k);
                v2f b = *(const v2f*)(br + k);
                acc = __builtin_amdgcn_wmma_f32_16x16x4_f32(false, a, false, b, (short)0,
                                                            acc, false, false);
            }
            float ac = att[o0 + l];
            #pragma unroll
            for (int v = 0; v < 8; v++) {
                float h = acc[v] + xl[(long)sIdx[v] * Dd + o0 + l]
                                 + xr[(long)tIdx[v] * Dd + o0 + l];
                h = (h > 0.0f) ? h : 0.2f * h;
                part[v] += h * ac;
            }
        }
        #pragma unroll
        for (int v = 0; v < 8; v++) {
            float pv = part[v];
            pv += __shfl_xor(pv, 8);
            pv += __shfl_xor(pv, 4);
            pv += __shfl_xor(pv, 2);
            pv += __shfl_xor(pv, 1);
            if (l == 0) logits[m0 + v + 8 * half] = pv;
        }
    }
}

// ---------------------------------------------------------------------------
// Scatter softmax pieces
__global__ void segmax_kernel(const float* __restrict__ logits, const int* __restrict__ eidx,
                              float* __restrict__ mmax) {
    long i = (long)blockIdx.x * blockDim.x + threadIdx.x;
    if (i < ENe) {
        int t = (i < Ee) ? eidx[Ee + i] : (int)(i - Ee);
        atomicMaxFloat(&mmax[t], logits[i]);
    }
}

__global__ void exdenom_kernel(const float* __restrict__ logits, const int* __restrict__ eidx,
                               const float* __restrict__ mmax, float* __restrict__ ex,
                               float* __restrict__ denom) {
    long i = (long)blockIdx.x * blockDim.x + threadIdx.x;
    if (i < ENe) {
        int t = (i < Ee) ? eidx[Ee + i] : (int)(i - Ee);
        float v = __expf(logits[i] - mmax[t]);
        ex[i] = v;
        atomicAdd(&denom[t], v);
    }
}

__global__ void alpha_scatter_kernel(const float* __restrict__ ex, const float* __restrict__ denom,
                                     const int* __restrict__ eidx, const float* __restrict__ xl,
                                     float* __restrict__ outc) {
    long i = blockIdx.x;   // one block per (edge or loop)
    int d = threadIdx.x;   // 128
    int s, t;
    if (i < Ee) { s = eidx[i]; t = eidx[Ee + i]; }
    else        { s = (int)(i - Ee); t = s; }
    float alpha = ex[i] / denom[t];
    atomicAdd(&outc[(long)t * Dd + d], alpha * xl[(long)s * Dd + d]);
}

// ---------------------------------------------------------------------------
// Mean pool over graphs (adds conv bias on the way)
__global__ void pool_kernel(const float* __restrict__ outc, const float* __restrict__ bias_conv,
                            const int* __restrict__ batch, float* __restrict__ pool,
                            float* __restrict__ cnt) {
    int n = blockIdx.x;    // one block per node
    int d = threadIdx.x;   // 128
    int g = batch[n];
    atomicAdd(&pool[(long)g * Dd + d], outc[(long)n * Dd + d] + bias_conv[d]);
    if (d == 0) atomicAdd(&cnt[g], 1.0f);
}

// Adapter linear + LayerNorm -> residual (= xn)
__global__ void adapter_ln_kernel(const float* __restrict__ pool, const float* __restrict__ cnt,
                                  const float* __restrict__ Wad, const float* __restrict__ bad,
                                  const float* __restrict__ lng, const float* __restrict__ lnb,
                                  float* __restrict__ resid) {
    int g = blockIdx.x, d = threadIdx.x;   // 128 threads
    __shared__ float row[Dd];
    __shared__ float red[Dd];
    float c = fmaxf(cnt[g], 1.0f);
    row[d] = pool[(long)g * Dd + d] / c;
    __syncthreads();
    float a = bad[d];
    for (int k = 0; k < Dd; k++) a += row[k] * Wad[(long)d * Dd + k];
    red[d] = a; __syncthreads();
    for (int s = 64; s > 0; s >>= 1) { if (d < s) red[d] += red[d + s]; __syncthreads(); }
    float mu = red[0] / (float)Dd;
    __syncthreads();
    float dm = a - mu;
    red[d] = dm * dm; __syncthreads();
    for (int s = 64; s > 0; s >>= 1) { if (d < s) red[d] += red[d + s]; __syncthreads(); }
    float var = red[0] / (float)Dd;
    resid[(long)g * Dd + d] = dm * rsqrtf(var + 1e-5f) * lng[d] + lnb[d];
}

// ---------------------------------------------------------------------------
// One residual-VQ step for quantizer q (launched sequentially q=0..2)
__global__ void rvq_step_kernel(const float* __restrict__ cb,   // [K, D] for this q
                                float* __restrict__ resid,      // [G, D] in/out
                                float* __restrict__ codes,      // [G, Q, D]
                                float* __restrict__ outIdx,     // [G, Q] as float
                                float* __restrict__ lossacc,    // scalar for this q
                                int q) {
    int g = blockIdx.x, d = threadIdx.x;   // 128 threads
    __shared__ float r[Dd];
    __shared__ float bd[Dd];
    __shared__ int   bi[Dd];
    r[d] = resid[(long)g * Dd + d];
    __syncthreads();
    float best = 3.0e38f; int bestk = 0;
    for (int kk = d; kk < Kk; kk += Dd) {
        const float* C = cb + (long)kk * Dd;
        float dot = 0.0f, nrm = 0.0f;
        for (int j = 0; j < Dd; j++) { float c = C[j]; dot += r[j] * c; nrm += c * c; }
        float dist = nrm - 2.0f * dot;      // argmin-equivalent (||r||^2 constant)
        if (dist < best) { best = dist; bestk = kk; }
    }
    bd[d] = best; bi[d] = bestk;
    __syncthreads();
    for (int s = 64; s > 0; s >>= 1) {
        if (d < s) {
            if (bd[d + s] < bd[d] || (bd[d + s] == bd[d] && bi[d + s] < bi[d])) {
                bd[d] = bd[d + s]; bi[d] = bi[d + s];
            }
        }
        __syncthreads();
    }
    int k0 = bi[0];
    __syncthreads();
    float c = cb[(long)k0 * Dd + d];
    float diff = c - r[d];
    codes[((long)g * Qq + q) * Dd + d] = c;
    resid[(long)g * Dd + d] = r[d] - c;
    bd[d] = diff * diff;
    __syncthreads();
    for (int s = 64; s > 0; s >>= 1) { if (d < s) bd[d] += bd[d + s]; __syncthreads(); }
    if (d == 0) {
        atomicAdd(lossacc, bd[0]);
        outIdx[(long)g * Qq + q] = (float)k0;
    }
}

__global__ void loss_final_kernel(const float* __restrict__ lossacc, float* __restrict__ out) {
    int q = threadIdx.x;
    if (q < Qq) out[q] = lossacc[q] / (float)(Gg * Dd);
}

// ---------------------------------------------------------------------------
extern "C" void kernel_launch(void* const* d_in, const int* in_sizes, int n_in,
                              void* d_out, int out_size, void* d_ws, size_t ws_size,
                              hipStream_t stream) {
    (void)in_sizes; (void)n_in; (void)out_size; (void)ws_size;
    const float* x         = (const float*)d_in[0];
    const int*   eidx      = (const int*)  d_in[1];
    const float* eattr     = (const float*)d_in[2];
    const int*   batch     = (const int*)  d_in[3];
    const float* W_l       = (const float*)d_in[4];
    const float* b_l       = (const float*)d_in[5];
    const float* W_r       = (const float*)d_in[6];
    const float* b_r       = (const float*)d_in[7];
    const float* W_e       = (const float*)d_in[8];
    const float* att       = (const float*)d_in[9];
    const float* bias_conv = (const float*)d_in[10];
    const float* W_ad      = (const float*)d_in[11];
    const float* b_ad      = (const float*)d_in[12];
    const float* ln_g      = (const float*)d_in[13];
    const float* ln_b      = (const float*)d_in[14];
    const float* codebooks = (const float*)d_in[15];
    const float* W_out     = (const float*)d_in[16];
    const float* b_out     = (const float*)d_in[17];
    float* out = (float*)d_out;

    // workspace carve-up (floats)
    float* ws = (float*)d_ws;
    size_t off = 0;
    float* xl      = ws + off; off += (size_t)Nn * Dd;
    float* xr      = ws + off; off += (size_t)Nn * Dd;
    float* lattr   = ws + off; off += (size_t)Nn * Dd;
    float* outc    = ws + off; off += (size_t)Nn * Dd;
    float* logits  = ws + off; off += (size_t)ENe;
    float* ex      = ws + off; off += (size_t)ENe;
    float* deg     = ws + off; off += (size_t)Nn;
    float* mmax    = ws + off; off += (size_t)Nn;
    float* denom   = ws + off; off += (size_t)Nn;
    float* pool    = ws + off; off += (size_t)Gg * Dd;
    float* cnt     = ws + off; off += (size_t)Gg;
    float* resid   = ws + off; off += (size_t)Gg * Dd;
    float* codes   = ws + off; off += (size_t)Gg * Qq * Dd;
    float* lossacc = ws + off; off += (size_t)Qq;

    float* out_q   = out;                           // [G,Q,FOUT]
    float* out_idx = out + (size_t)Gg * Qq * FOUTc; // [G,Q]
    float* out_ls  = out_idx + (size_t)Gg * Qq;     // [Q]

    // zero-init accumulators (must run every call)
    (void)hipMemsetAsync(lattr,   0, (size_t)Nn * Dd * sizeof(float), stream);
    (void)hipMemsetAsync(outc,    0, (size_t)Nn * Dd * sizeof(float), stream);
    (void)hipMemsetAsync(deg,     0, (size_t)Nn * sizeof(float), stream);
    (void)hipMemsetAsync(denom,   0, (size_t)Nn * sizeof(float), stream);
    (void)hipMemsetAsync(pool,    0, (size_t)Gg * Dd * sizeof(float), stream);
    (void)hipMemsetAsync(cnt,     0, (size_t)Gg * sizeof(float), stream);
    (void)hipMemsetAsync(lossacc, 0, (size_t)Qq * sizeof(float), stream);
    fill_neginf_kernel<<<(Nn + 255) / 256, 256, 0, stream>>>(mmax, Nn);

    // self-loop edge_attr mean
    deg_loop_kernel<<<Ee, Dd, 0, stream>>>(eidx, eattr, lattr, deg);
    loop_norm_kernel<<<((size_t)Nn * Dd + 255) / 256, 256, 0, stream>>>(lattr, deg);

    // node transforms via WMMA
    gemm128_wmma<<<dim3(Nn / 16, Dd / 16), 32, 0, stream>>>(x, W_l, b_l, xl, Dd);
    gemm128_wmma<<<dim3(Nn / 16, Dd / 16), 32, 0, stream>>>(x, W_r, b_r, xr, Dd);

    // fused edge GEMM + gather + attention logits (persistent grid-stride)
    edge_logits_wmma<<<2720, 128, 0, stream>>>(eattr, lattr, eidx, W_e, att,
                                               xl, xr, logits);

    // scatter softmax + aggregation
    segmax_kernel<<<(ENe + 255) / 256, 256, 0, stream>>>(logits, eidx, mmax);
    exdenom_kernel<<<(ENe + 255) / 256, 256, 0, stream>>>(logits, eidx, mmax, ex, denom);
    alpha_scatter_kernel<<<ENe, Dd, 0, stream>>>(ex, denom, eidx, xl, outc);

    // pool + adapter + LN
    pool_kernel<<<Nn, Dd, 0, stream>>>(outc, bias_conv, batch, pool, cnt);
    adapter_ln_kernel<<<Gg, Dd, 0, stream>>>(pool, cnt, W_ad, b_ad, ln_g, ln_b, resid);

    // residual VQ (sequential over quantizers)
    for (int q = 0; q < Qq; q++) {
        rvq_step_kernel<<<Gg, Dd, 0, stream>>>(codebooks + (size_t)q * Kk * Dd,
                                               resid, codes, out_idx, lossacc + q, q);
    }
    loss_final_kernel<<<1, Qq, 0, stream>>>(lossacc, out_ls);

    // output projection via WMMA: [G*Q,128] @ [FOUT,128]^T + b_out -> d_out
    gemm128_wmma<<<dim3((Gg * Qq) / 16, FOUTc / 16), 32, 0, stream>>>(codes, W_out, b_out,
                                                                      out_q, FOUTc);
}